// QuantumLayer_71957882077757
// MI455X (gfx1250) — compile-verified
//
#include <hip/hip_runtime.h>
#include <hip/hip_bf16.h>

#define N_QUBITS 13
#define DIM      (1 << N_QUBITS)   // 8192 amplitudes
#define N_LAYERS 2
#define BATCH    512
#define TPB      256               // 8 wave32 waves

typedef float v2f __attribute__((ext_vector_type(2)));
typedef float v8f __attribute__((ext_vector_type(8)));

__device__ __forceinline__ float2 cmul(float2 a, float2 b) {
    return make_float2(a.x * b.x - a.y * b.y, a.x * b.y + a.y * b.x);
}
__device__ __forceinline__ float2 cadd(float2 a, float2 b) {
    return make_float2(a.x + b.x, a.y + b.y);
}

// One workgroup per batch element; full state vector resident in LDS.
// Ping-pong buffers (128 KB of the WGP's 320 KB LDS) let the whole
// 12-CNOT chain collapse into ONE permutation pass per layer.
__global__ __launch_bounds__(TPB)
void qsim_kernel(const float* __restrict__ x,      // (BATCH, 13)
                 const float* __restrict__ w,      // (2, 13, 3)
                 float* __restrict__ probs,        // (BATCH, DIM) or null
                 float* __restrict__ out)          // (BATCH, 13) when probs==null
{
    __shared__ float2 lds[2][DIM];                 // 2 x 64 KB ping-pong state
    const int b   = blockIdx.x;
    const int tid = threadIdx.x;

    // |00...0>
    for (int s = tid; s < DIM; s += TPB)
        lds[0][s] = make_float2(s == 0 ? 1.0f : 0.0f, 0.0f);
    __syncthreads();

    int cur = 0;
    for (int l = 0; l < N_LAYERS; ++l) {
        for (int q = 0; q < N_QUBITS; ++q) {
            // Fused U = RZ(w2) * RY(x*w1) * RX(x*w0); uniform across the block.
            const float xi = x[b * N_QUBITS + q];
            const float w0 = w[(l * N_QUBITS + q) * 3 + 0];
            const float w1 = w[(l * N_QUBITS + q) * 3 + 1];
            const float w2 = w[(l * N_QUBITS + q) * 3 + 2];
            const float tx = 0.5f * xi * w0, ty = 0.5f * xi * w1, tz = 0.5f * w2;
            const float cx = cosf(tx), sx = sinf(tx);
            const float cy = cosf(ty), sy = sinf(ty);
            // Rx = [[cx, -i sx], [-i sx, cx]],  Ry = [[cy, -sy], [sy, cy]]
            float2 Rx00 = { cx, 0.f }, Rx01 = { 0.f, -sx };
            float2 Rx10 = { 0.f, -sx }, Rx11 = { cx, 0.f };
            float2 Ry00 = { cy, 0.f }, Ry01 = { -sy, 0.f };
            float2 Ry10 = { sy, 0.f }, Ry11 = { cy, 0.f };
            float2 M00 = cadd(cmul(Ry00, Rx00), cmul(Ry01, Rx10));
            float2 M01 = cadd(cmul(Ry00, Rx01), cmul(Ry01, Rx11));
            float2 M10 = cadd(cmul(Ry10, Rx00), cmul(Ry11, Rx10));
            float2 M11 = cadd(cmul(Ry10, Rx01), cmul(Ry11, Rx11));
            const float cz = cosf(tz), sz = sinf(tz);
            const float2 em = { cz, -sz }, ep = { cz, sz };   // RZ = diag(e^{-i p/2}, e^{+i p/2})
            const float2 U00 = cmul(em, M00), U01 = cmul(em, M01);
            const float2 U10 = cmul(ep, M10), U11 = cmul(ep, M11);

            const int pbit = (N_QUBITS - 1) - q;              // wire 0 = MSB
            const int strd = 1 << pbit;
            float2* S = lds[cur];
            for (int t = tid; t < (DIM >> 1); t += TPB) {
                const int lo = t & (strd - 1);
                const int i0 = ((t >> pbit) << (pbit + 1)) | lo;
                const int i1 = i0 | strd;
                const float2 a0 = S[i0], a1 = S[i1];
                S[i0] = cadd(cmul(U00, a0), cmul(U01, a1));
                S[i1] = cadd(cmul(U10, a0), cmul(U11, a1));
            }
            __syncthreads();
        }
        // Full CNOT chain == basis permutation: dest MSB-first bit j is the
        // prefix-XOR of source wire bits 0..j. One pass, one barrier.
        float2* S   = lds[cur];
        float2* Dst = lds[cur ^ 1];
        for (int s = tid; s < DIM; s += TPB) {
            unsigned d = 0, par = 0;
            #pragma unroll
            for (int qb = N_QUBITS - 1; qb >= 0; --qb) {
                par ^= ((unsigned)s >> qb) & 1u;
                d   |= par << qb;
            }
            Dst[d] = S[s];
        }
        cur ^= 1;
        __syncthreads();
    }

    float2* S = lds[cur];                                     // cur == 0 after 2 layers
    if (probs) {
        float* P = probs + (size_t)b * DIM;
        for (int s = tid; s < DIM; s += TPB) {
            const float2 a = S[s];
            P[s] = a.x * a.x + a.y * a.y;                     // coalesced b32 stores
        }
    } else {
        // Fallback: signed reduction in-kernel (scratch too small for probs)
        float acc[N_QUBITS];
        #pragma unroll
        for (int i = 0; i < N_QUBITS; ++i) acc[i] = 0.0f;
        for (int s = tid; s < DIM; s += TPB) {
            const float2 a = S[s];
            const float p = a.x * a.x + a.y * a.y;
            #pragma unroll
            for (int i = 0; i < N_QUBITS; ++i)
                acc[i] += ((s >> ((N_QUBITS - 1) - i)) & 1) ? -p : p;
        }
        float* sums = (float*)lds[cur ^ 1];                   // free 64 KB buffer
        #pragma unroll
        for (int i = 0; i < N_QUBITS; ++i) sums[tid * 16 + i] = acc[i];
        __syncthreads();
        for (int off = TPB / 2; off > 0; off >>= 1) {
            if (tid < off)
                for (int i = 0; i < N_QUBITS; ++i)
                    sums[tid * 16 + i] += sums[(tid + off) * 16 + i];
            __syncthreads();
        }
        if (tid < N_QUBITS) out[b * N_QUBITS + tid] = sums[tid];
    }
}

// E(512x13) = P(512x8192) x Sign(8192x16-padded) via v_wmma_f32_16x16x4_f32.
// A 16x4 f32 layout (ISA 7.12.2): lanes 0-15 hold K={0,1} in VGPR{0,1},
// lanes 16-31 hold K={2,3}; row M = lane&15. B rows striped across lanes
// per VGPR (VGPR0: K=k,k+2; VGPR1: K=k+1,k+3; col N = lane&15).
// Sign[s][n] = 1 - 2*bit_{12-n}(s) for n<13, 0 otherwise — built in registers.
__global__ __launch_bounds__(256)
void expval_wmma_kernel(const float* __restrict__ probs, float* __restrict__ out)
{
    __shared__ float red[8 * 256];           // 8 wave-partial 16x16 tiles
    const int tid  = threadIdx.x;
    const int wave = tid >> 5;
    const int lane = tid & 31;
    const int m    = lane & 15;
    const int hi   = lane >> 4;
    const int rowBase = blockIdx.x * 16;

    const float* Prow = probs + (size_t)(rowBase + m) * DIM;
    v8f acc = {};
    const int k0 = wave * (DIM / 8);         // 8 waves split K=8192
    const int k1 = k0 + (DIM / 8);
    for (int k = k0; k < k1; k += 4) {
        const int ks = k + 2 * hi;
        v2f a;
        a.x = Prow[ks];
        a.y = Prow[ks + 1];
        v2f bm;
        bm.x = (m < N_QUBITS) ? (1.0f - 2.0f * (float)((ks       >> ((N_QUBITS - 1) - m)) & 1)) : 0.0f;
        bm.y = (m < N_QUBITS) ? (1.0f - 2.0f * (float)(((ks + 1) >> ((N_QUBITS - 1) - m)) & 1)) : 0.0f;
        acc = __builtin_amdgcn_wmma_f32_16x16x4_f32(
            /*neg_a=*/false, a, /*neg_b=*/false, bm,
            /*c_mod=*/(short)0, acc, /*reuse_a=*/false, /*reuse_b=*/false);
    }
    // C/D layout: VGPR r -> (M=r, N=lane) lanes 0-15, (M=r+8, N=lane-16) lanes 16-31
    #pragma unroll
    for (int r = 0; r < 8; ++r)
        red[wave * 256 + (r + 8 * hi) * 16 + m] = acc[r];
    __syncthreads();

    float s = 0.0f;
    #pragma unroll
    for (int wv = 0; wv < 8; ++wv) s += red[wv * 256 + tid];
    const int row = tid >> 4, col = tid & 15;
    if (col < N_QUBITS)
        out[(rowBase + row) * N_QUBITS + col] = s;
}

extern "C" void kernel_launch(void* const* d_in, const int* in_sizes, int n_in,
                              void* d_out, int out_size, void* d_ws, size_t ws_size,
                              hipStream_t stream) {
    const float* x = (const float*)d_in[0];   // (512, 13) f32
    const float* w = (const float*)d_in[1];   // (2, 13, 3) f32
    float* out = (float*)d_out;               // (512, 13) f32

    const size_t probs_bytes = (size_t)BATCH * DIM * sizeof(float);
    if (ws_size >= probs_bytes) {
        float* probs = (float*)d_ws;
        qsim_kernel<<<BATCH, TPB, 0, stream>>>(x, w, probs, nullptr);
        expval_wmma_kernel<<<BATCH / 16, 256, 0, stream>>>(probs, out);
    } else {
        qsim_kernel<<<BATCH, TPB, 0, stream>>>(x, w, nullptr, out);
    }
}